// HeteroLayer_81707457839501
// MI455X (gfx1250) — compile-verified
//
#include <hip/hip_runtime.h>
#include <hip/hip_bf16.h>

typedef float v2f __attribute__((ext_vector_type(2)));
typedef float v8f __attribute__((ext_vector_type(8)));

#define D 128
#define EPS 1e-5f

// ---------------------------------------------------------------------------
// Zero a float region (agg buffers + stats), grid-stride.
// ---------------------------------------------------------------------------
__global__ void zero_kernel(float* __restrict__ p, size_t n) {
    size_t i = (size_t)blockIdx.x * blockDim.x + threadIdx.x;
    size_t stride = (size_t)gridDim.x * blockDim.x;
    for (; i < n; i += stride) p[i] = 0.0f;
}

// ---------------------------------------------------------------------------
// wroot_var = W_root_adj + W_root_groups ; bias_var = b_adj + b_groups
// ---------------------------------------------------------------------------
__global__ void combine_kernel(const float* __restrict__ wa, const float* __restrict__ wg,
                               const float* __restrict__ ba, const float* __restrict__ bg,
                               float* __restrict__ wout, float* __restrict__ bout) {
    int i = blockIdx.x * blockDim.x + threadIdx.x;
    if (i < D * D) wout[i] = wa[i] + wg[i];
    else if (i < D * D + D) { int j = i - D * D; bout[j] = ba[j] + bg[j]; }
}

// ---------------------------------------------------------------------------
// Edge scatter-add: one wave32 per edge, lane handles float4 (32*16B = 512B row)
// agg[dst[e]][:] += xsrc[src[e]][:]
// ---------------------------------------------------------------------------
__global__ void scatter_kernel(const float* __restrict__ xsrc,
                               const int* __restrict__ src, const int* __restrict__ dst,
                               float* __restrict__ agg, int E) {
    int lane = threadIdx.x & 31;
    long wave = (long)blockIdx.x * (blockDim.x >> 5) + (threadIdx.x >> 5);
    long nwaves = (long)gridDim.x * (blockDim.x >> 5);
    for (long e = wave; e < E; e += nwaves) {
        int s = src[e];
        int d = dst[e];
        const float4 v = *(const float4*)(xsrc + (size_t)s * D + lane * 4);
        float* p = agg + (size_t)d * D + lane * 4;
        atomicAdd(p + 0, v.x);
        atomicAdd(p + 1, v.y);
        atomicAdd(p + 2, v.z);
        atomicAdd(p + 3, v.w);
    }
}

// ---------------------------------------------------------------------------
// Fused multi-input GEMM: out = sum_p A_p @ W_p + bias, plus column sum/sumsq
// accumulation for GraphNorm.  M % 16 == 0 assumed (holds: 200000, 100000).
// Block = 256 threads = 8 waves; wave w computes rows [blk*128 + 16w, +16),
// all 128 output columns, via V_WMMA_F32_16X16X4_F32 (8 N-tiles per K-step).
// Weight tile staged through LDS in 64-row chunks (32 KB).
// ---------------------------------------------------------------------------
__global__ void __launch_bounds__(256)
gemm3_stats_kernel(const float* __restrict__ A0, const float* __restrict__ W0,
                   const float* __restrict__ A1, const float* __restrict__ W1,
                   const float* __restrict__ A2, const float* __restrict__ W2,
                   const float* __restrict__ bias,
                   float* __restrict__ out, int M,
                   float* __restrict__ colsum, float* __restrict__ colsumsq) {
    __shared__ float sW[64 * D];     // 32 KB weight chunk
    __shared__ float sSum[D];
    __shared__ float sSq[D];

    const int tid   = threadIdx.x;
    const int lane  = tid & 31;
    const int wave  = tid >> 5;
    const int laneM = lane & 15;     // row (A) / col (B,C) selector within halves
    const int kHi   = lane >> 4;     // 0: K pair (k,k+1); 1: K pair (k+2,k+3)
    const int r0    = blockIdx.x * 128 + wave * 16;
    const bool valid = (r0 < M);

    v8f acc[8];
#pragma unroll
    for (int nt = 0; nt < 8; ++nt) acc[nt] = (v8f)(0.0f);

    const float* As[3] = {A0, A1, A2};
    const float* Ws[3] = {W0, W1, W2};

    for (int p = 0; p < 3; ++p) {
        if (As[p] == nullptr) continue;          // uniform across the block
        const float* A = As[p];
        const float* W = Ws[p];
        const float* arow = A + (size_t)(r0 + laneM) * D + 2 * kHi;

        for (int chunk = 0; chunk < 2; ++chunk) {
            const int c0 = chunk * 64;
            __syncthreads();                      // protect previous sW use
            for (int i = tid; i < 64 * D; i += 256) sW[i] = W[c0 * D + i];
            __syncthreads();

            if (valid) {
#pragma unroll 4
                for (int k = 0; k < 64; k += 4) {
                    // A fragment: lanes 0-15 -> A[M=laneM][k..k+1],
                    //             lanes 16-31 -> A[M=laneM][k+2..k+3]
                    v2f a = *(const v2f*)(arow + c0 + k);
#pragma unroll
                    for (int nt = 0; nt < 8; ++nt) {
                        // B fragment: lanes 0-15 -> W[k..k+1][n],
                        //             lanes 16-31 -> W[k+2..k+3][n]
                        const int n = nt * 16 + laneM;
                        v2f b;
                        b.x = sW[(k + 2 * kHi) * D + n];
                        b.y = sW[(k + 2 * kHi + 1) * D + n];
                        acc[nt] = __builtin_amdgcn_wmma_f32_16x16x4_f32(
                            false, a, false, b, (short)0, acc[nt], false, false);
                    }
                }
            }
        }
    }

    // ----- epilogue: bias, store, per-column stats -----
    if (tid < D) { sSum[tid] = 0.0f; sSq[tid] = 0.0f; }
    __syncthreads();

    if (valid) {
#pragma unroll
        for (int nt = 0; nt < 8; ++nt) {
            const int col = nt * 16 + laneM;
            const float bs = bias[col];
            float lsum = 0.0f, lsq = 0.0f;
#pragma unroll
            for (int r = 0; r < 8; ++r) {
                // C/D layout: VGPR r, lanes 0-15 -> M=r, lanes 16-31 -> M=8+r
                const float v = acc[nt][r] + bs;
                const int row = r0 + r + 8 * kHi;
                out[(size_t)row * D + col] = v;
                lsum += v;
                lsq  += v * v;
            }
            atomicAdd(&sSum[col], lsum);
            atomicAdd(&sSq[col], lsq);
        }
    }
    __syncthreads();
    if (tid < D) {
        atomicAdd(&colsum[tid], sSum[tid]);
        atomicAdd(&colsumsq[tid], sSq[tid]);
    }
}

// ---------------------------------------------------------------------------
// GraphNorm + ReLU, in place:  relu(w * (x - ms*mu) * rsqrt(var+eps) + b)
// var = E[x^2] - 2*ms*mu*E[x] + (ms*mu)^2   (stats from colsum / colsumsq)
// ---------------------------------------------------------------------------
__global__ void norm_relu_kernel(float* __restrict__ x, int M,
                                 const float* __restrict__ w, const float* __restrict__ b,
                                 const float* __restrict__ ms,
                                 const float* __restrict__ colsum,
                                 const float* __restrict__ colsumsq) {
    const float invM = 1.0f / (float)M;
    size_t n = (size_t)M * D;
    size_t i = (size_t)blockIdx.x * blockDim.x + threadIdx.x;
    size_t stride = (size_t)gridDim.x * blockDim.x;
    for (; i < n; i += stride) {
        int col = (int)(i & (D - 1));
        float mu = colsum[col] * invM;
        float m  = ms[col] * mu;
        float var = colsumsq[col] * invM - 2.0f * m * mu + m * m;
        float inv = rsqrtf(var + EPS);
        float v = (x[i] - m) * inv * w[col] + b[col];
        x[i] = v > 0.0f ? v : 0.0f;
    }
}

// ---------------------------------------------------------------------------
extern "C" void kernel_launch(void* const* d_in, const int* in_sizes, int n_in,
                              void* d_out, int out_size, void* d_ws, size_t ws_size,
                              hipStream_t stream) {
    const float* x_var       = (const float*)d_in[0];
    const float* x_con       = (const float*)d_in[1];
    const float* x_region    = (const float*)d_in[2];
    const int*   src_adj     = (const int*)d_in[3];
    const int*   dst_adj     = (const int*)d_in[4];
    const int*   src_touch   = (const int*)d_in[5];
    const int*   dst_touch   = (const int*)d_in[6];
    const int*   src_groups  = (const int*)d_in[7];
    const int*   dst_groups  = (const int*)d_in[8];
    const float* W_rel_adj   = (const float*)d_in[9];
    const float* b_adj       = (const float*)d_in[10];
    const float* W_root_adj  = (const float*)d_in[11];
    const float* W_rel_tch   = (const float*)d_in[12];
    const float* b_tch       = (const float*)d_in[13];
    const float* W_root_tch  = (const float*)d_in[14];
    const float* W_rel_grp   = (const float*)d_in[15];
    const float* b_grp       = (const float*)d_in[16];
    const float* W_root_grp  = (const float*)d_in[17];
    const float* gn_w_var    = (const float*)d_in[18];
    const float* gn_b_var    = (const float*)d_in[19];
    const float* gn_ms_var   = (const float*)d_in[20];
    const float* gn_w_con    = (const float*)d_in[21];
    const float* gn_b_con    = (const float*)d_in[22];
    const float* gn_ms_con   = (const float*)d_in[23];

    const int NVAR = in_sizes[0] / D;
    const int NCON = in_sizes[1] / D;
    const int E_adj = in_sizes[3];
    const int E_tch = in_sizes[5];
    const int E_grp = in_sizes[7];

    // ---- workspace layout (floats) ----
    float* ws = (float*)d_ws;
    size_t off = 0;
    float* agg_adj  = ws + off; off += (size_t)NVAR * D;
    float* agg_grp  = ws + off; off += (size_t)NVAR * D;
    float* agg_con  = ws + off; off += (size_t)NCON * D;
    float* wroot_v  = ws + off; off += (size_t)D * D;
    float* bias_v   = ws + off; off += D;
    float* s_var    = ws + off;
    float* q_var    = s_var + D;
    float* s_con    = s_var + 2 * D;
    float* q_con    = s_var + 3 * D;
    off += 4 * D;
    const size_t zeroN = off;

    float* out_var = (float*)d_out;
    float* out_con = out_var + (size_t)NVAR * D;

    // 1) clear aggregation buffers + stats (ws is poisoned by harness)
    zero_kernel<<<2048, 256, 0, stream>>>(ws, zeroN);

    // 2) pre-sum the two var-side root weights / biases
    combine_kernel<<<(D * D + D + 255) / 256, 256, 0, stream>>>(
        W_root_adj, W_root_grp, b_adj, b_grp, wroot_v, bias_v);

    // 3) edge scatter-adds (memory-bound portion)
    scatter_kernel<<<2048, 256, 0, stream>>>(x_var,    src_adj,    dst_adj,    agg_adj, E_adj);
    scatter_kernel<<<2048, 256, 0, stream>>>(x_region, src_groups, dst_groups, agg_grp, E_grp);
    scatter_kernel<<<2048, 256, 0, stream>>>(x_var,    src_touch,  dst_touch,  agg_con, E_tch);

    // 4) fused GEMMs (WMMA f32) + column stats
    gemm3_stats_kernel<<<(NVAR + 127) / 128, 256, 0, stream>>>(
        agg_adj, W_rel_adj, agg_grp, W_rel_grp, x_var, wroot_v,
        bias_v, out_var, NVAR, s_var, q_var);
    gemm3_stats_kernel<<<(NCON + 127) / 128, 256, 0, stream>>>(
        agg_con, W_rel_tch, x_con, W_root_tch, nullptr, nullptr,
        b_tch, out_con, NCON, s_con, q_con);

    // 5) GraphNorm + ReLU (in place on d_out)
    norm_relu_kernel<<<2048, 256, 0, stream>>>(out_var, NVAR, gn_w_var, gn_b_var, gn_ms_var, s_var, q_var);
    norm_relu_kernel<<<2048, 256, 0, stream>>>(out_con, NCON, gn_w_con, gn_b_con, gn_ms_con, s_con, q_con);
}